// StraightThroughEstimator_62938450755771
// MI455X (gfx1250) — compile-verified
//
#include <hip/hip_runtime.h>
#include <math.h>

// Problem shape from the reference setup (fixed): L=5 layers, A=15 alpha
// boundaries/layer, R=40 r boundaries/layer, points = 128*30000.
#define NLAYER 5
#define NA 15
#define NR 40

// LDS float layout (dynamic shared memory, offsets in floats).
// Strides are odd (17 / 67 / 41) so the 5 layers occupy distinct LDS banks
// ((stride*layer + x) mod 64 differs per layer), while the binary-search
// window stays a power of two (16 / 64) via +inf padding.
#define STR_A  17   // 15 real + 1 inf pad + 1 dead slot
#define STR_R  67   // 40 real + 24 inf pads (search window 64) + 3 dead
#define OFF_ABP  0                       // alpha boundaries, padded  5*17 = 85
#define OFF_RBLP (OFF_ABP + NLAYER*STR_A)   // r log boundaries, padded 5*67 = 335
#define OFF_AT   (OFF_RBLP + NLAYER*STR_R)  // alpha midpoint LUT 5*17 = 85
#define OFF_RT   (OFF_AT + NLAYER*STR_A)    // transformed r LUT 5*41 = 205
#define SMEM_FLOATS (OFF_RT + NLAYER*(NR+1))

struct F3 { float a, b, c; };   // 12-byte point record -> b96 load/store path

__global__ __launch_bounds__(256) void ste_gfx1250_kernel(
    const float* __restrict__ x,
    const float* __restrict__ zb,
    const float* __restrict__ ab,
    const float* __restrict__ rb,
    const float* __restrict__ rbl,
    const float* __restrict__ rabs,
    const float* __restrict__ fmax_,
    const float* __restrict__ fmin_,
    const float* __restrict__ fnorm,
    const float* __restrict__ fshift,
    float* __restrict__ out,
    int npts)
{
  extern __shared__ float smem[];
  const int tid = threadIdx.x;

  float* s_abp  = smem + OFF_ABP;
  float* s_rblp = smem + OFF_RBLP;
  float* s_at   = smem + OFF_AT;
  float* s_rt   = smem + OFF_RT;

  const float INF = __builtin_huge_valf();

  // ---- Stage 1a: async global->LDS staging of the searched boundary arrays
  // into their padded slots (CDNA5 async path, tracked by ASYNCcnt).
  {
    const unsigned ldsBase = __builtin_amdgcn_groupstaticsize(); // dyn-LDS start
    for (int idx = tid; idx < NLAYER * NA; idx += 256) {         // 75 alpha
      int layer = idx / NA, j = idx % NA;
      unsigned dst = ldsBase + (unsigned)(OFF_ABP + layer * STR_A + j) * 4u;
      const float* src = ab + idx;
      asm volatile("global_load_async_to_lds_b32 %0, %1, off"
                   :: "v"(dst), "v"(src) : "memory");
    }
    for (int idx = tid; idx < NLAYER * NR; idx += 256) {         // 200 r-log
      int layer = idx / NR, j = idx % NR;
      unsigned dst = ldsBase + (unsigned)(OFF_RBLP + layer * STR_R + j) * 4u;
      const float* src = rbl + idx;
      asm volatile("global_load_async_to_lds_b32 %0, %1, off"
                   :: "v"(dst), "v"(src) : "memory");
    }
    asm volatile("s_wait_asynccnt 0x0" ::: "memory");
  }
  // ---- Stage 1b: +inf pads (plain ds stores, disjoint addresses)
  for (int idx = tid; idx < NLAYER * 2; idx += 256)              // alpha pads
    s_abp[(idx >> 1) * STR_A + NA + (idx & 1)] = INF;
  for (int idx = tid; idx < NLAYER * (STR_R - NR); idx += 256) { // r pads
    int layer = idx / (STR_R - NR), j = NR + idx % (STR_R - NR);
    s_rblp[layer * STR_R + j] = INF;
  }
  __syncthreads();

  // ---- Scalars (uniform)
  const float shift0 = fshift[0];
  const float shift2 = fshift[2];
  const float norm2  = fnorm[2];
  const float min2   = fmin_[2];
  const float max2   = fmax_[2];
  const float minR   = 0.5f * (rabs[1 * NR + 0]      + rabs[1 * NR + 1]);
  const float maxR   = 0.5f * (rabs[4 * NR + NR - 1] + rabs[4 * NR + NR - 2]);
  const float lo = shift0, hi = 1.0f + shift0;

  // z boundaries + z midpoint table live entirely in registers (4 floats)
  const float zb0 = zb[0], zb1 = zb[1], zb2 = zb[2], zb3 = zb[3];
  const float zt1 = 0.5f * (zb0 + zb1);
  const float zt2 = 0.5f * (zb1 + zb2);
  const float zt3 = 0.5f * (zb2 + zb3);

  // ---- Stage 2: derived LUTs (alpha midpoints; fully-transformed r values:
  // mid -> unnormalize(minR,maxR) -> log(.+1e-10) -> normalize -> clip).
  for (int idx = tid; idx < NLAYER * 16 + NLAYER * (NR + 1); idx += 256) {
    if (idx < NLAYER * 16) {                         // alpha LUT 5x16
      int layer = idx >> 4, j = idx & 15;
      const float* b = s_abp + layer * STR_A;
      float v = (j == 0) ? lo : (j == NA) ? hi : 0.5f * (b[j] + b[j - 1]);
      s_at[layer * STR_A + j] = v;
    } else {                                         // r LUT 5x41
      int k = idx - NLAYER * 16, layer = k / (NR + 1), j = k % (NR + 1);
      const float* b = rb + layer * NR;              // NON-log boundaries
      float rv = (j == 0) ? lo : (j == NR) ? hi : 0.5f * (b[j] + b[j - 1]);
      float un = (rv - shift2) / norm2 * maxR + minR;
      float rn = (logf(un + 1e-10f) - min2) / max2 * norm2 + shift2;
      rn = fminf(fmaxf(rn, lo), hi);
      s_rt[layer * (NR + 1) + j] = rn;
    }
  }
  __syncthreads();

  // ---- Stage 3: streaming loop. b96 load, register z-search, branchless
  // power-of-2 binary searches in LDS (4 probes alpha, 6 probes r), two LDS
  // LUT reads, b96 store. ~48 B LDS + 24 B HBM per point -> HBM bound.
  const F3* __restrict__ xv = (const F3*)x;
  F3* __restrict__ ov = (F3*)out;
  const int stride = (int)gridDim.x * 256;

  for (int p = (int)blockIdx.x * 256 + tid; p < npts; p += stride) {
    if (p + stride < npts)
      __builtin_prefetch(&xv[p + stride], 0, 0);     // global_prefetch_b8

    const F3 v = xv[p];

    // z: searchsorted(left) = count of boundaries < z (registers only)
    int zbin = (zb0 < v.a) + (zb1 < v.a) + (zb2 < v.a) + (zb3 < v.a);
    float zout = (zbin == 0) ? lo
               : (zbin == 1) ? zt1
               : (zbin == 2) ? zt2
               : (zbin == 3) ? zt3 : hi;

    // alpha: branchless lower-bound over 16 padded entries (4 LDS probes)
    const float* abL = s_abp + zbin * STR_A;
    int ai = 0;
    ai += (abL[ai + 7] < v.b) ? 8 : 0;
    ai += (abL[ai + 3] < v.b) ? 4 : 0;
    ai += (abL[ai + 1] < v.b) ? 2 : 0;
    ai += (abL[ai]     < v.b) ? 1 : 0;

    // r: branchless lower-bound over 64 padded entries (6 LDS probes)
    const float* rbL = s_rblp + zbin * STR_R;
    int ri = 0;
    ri += (rbL[ri + 31] < v.c) ? 32 : 0;
    ri += (rbL[ri + 15] < v.c) ? 16 : 0;
    ri += (rbL[ri + 7]  < v.c) ?  8 : 0;
    ri += (rbL[ri + 3]  < v.c) ?  4 : 0;
    ri += (rbL[ri + 1]  < v.c) ?  2 : 0;
    ri += (rbL[ri]      < v.c) ?  1 : 0;

    F3 o;
    o.a = zout;
    o.b = s_at[zbin * STR_A + ai];
    o.c = s_rt[zbin * (NR + 1) + ri];
    ov[p] = o;
  }
}

extern "C" void kernel_launch(void* const* d_in, const int* in_sizes, int n_in,
                              void* d_out, int out_size, void* d_ws, size_t ws_size,
                              hipStream_t stream) {
  const float* x      = (const float*)d_in[0];
  const float* zb     = (const float*)d_in[1];
  const float* ab     = (const float*)d_in[2];
  const float* rb     = (const float*)d_in[3];
  const float* rbl    = (const float*)d_in[4];
  const float* rabs   = (const float*)d_in[5];
  const float* fmax_  = (const float*)d_in[6];
  const float* fmin_  = (const float*)d_in[7];
  const float* fnorm  = (const float*)d_in[8];
  const float* fshift = (const float*)d_in[9];
  float* out = (float*)d_out;

  const int npts = in_sizes[0] / 3;            // 3,840,000
  int blocks = 2048;                           // grid-stride; amortizes setup
  int want = (npts + 255) / 256;
  if (blocks > want) blocks = want;
  if (blocks < 1) blocks = 1;

  ste_gfx1250_kernel<<<blocks, 256, SMEM_FLOATS * sizeof(float), stream>>>(
      x, zb, ab, rb, rbl, rabs, fmax_, fmin_, fnorm, fshift, out, npts);
}